// LIFSpike_71674414236194
// MI455X (gfx1250) — compile-verified
//
#include <hip/hip_runtime.h>

// LIF spiking neuron over T=4 timesteps.
// x layout: [T*B, C, H, W] fp32 = 4 planes of 8,388,608 floats (32 MiB each).
// Per spatial index i (independent across i):
//   mem = mem*TAU + x[t*NT + i]; spike = (mem >= THRESH); mem = spike ? 0 : mem
// Pure streaming workload: 128 MiB in, 128 MiB out -> HBM-bandwidth-bound.
// All indices fit in 32 bits (max 33,554,432 < 2^31): use int offsets so the
// compiler emits saddr(SGPR64)+voffset(VGPR32) addressing instead of 64-bit
// per-lane address chains.

typedef float float4v __attribute__((ext_vector_type(4)));

#define T_STEPS 4
#define TAU     0.5f
#define THRESH  1.0f

__global__ __launch_bounds__(256) void lif_stream_kernel(
    const float* __restrict__ x,
    float* __restrict__ out,
    int nvec_per_t)   // number of float4 vectors per timestep plane
{
    const int nt_elems = nvec_per_t * 4;          // floats per plane (8,388,608)
    const int stride   = gridDim.x * blockDim.x;

    for (int i = blockIdx.x * blockDim.x + threadIdx.x;
         i < nvec_per_t; i += stride)
    {
        const int base = i * 4;

        // Prefetch next grid-stride iteration's tiles into GL2
        // (global_prefetch_b8, speculative).
        const int inext = i + stride;
        if (inext < nvec_per_t) {
            const int bnext = inext * 4;
#pragma unroll
            for (int t = 0; t < T_STEPS; ++t) {
                __builtin_prefetch(x + t * nt_elems + bnext, 0, 0);
            }
        }

        // Issue all 4 plane loads up front (independent -> 4-deep MLP),
        // non-temporal: each byte is read exactly once.
        float4v xt[T_STEPS];
#pragma unroll
        for (int t = 0; t < T_STEPS; ++t) {
            xt[t] = __builtin_nontemporal_load(
                (const float4v*)(x + t * nt_elems + base));
        }

        // 4-step recurrence in registers, 4 components per thread.
        float4v mem = {0.0f, 0.0f, 0.0f, 0.0f};
#pragma unroll
        for (int t = 0; t < T_STEPS; ++t) {
            float4v sp;
#pragma unroll
            for (int c = 0; c < 4; ++c) {
                float m = __builtin_fmaf(mem[c], TAU, xt[t][c]); // leak+integrate
                bool fired = (m >= THRESH);
                sp[c]  = fired ? 1.0f : 0.0f;                    // fire
                mem[c] = fired ? 0.0f : m;                       // hard reset
            }
            // Streaming store, non-temporal: written once, never re-read.
            __builtin_nontemporal_store(
                sp, (float4v*)(out + t * nt_elems + base));
        }
    }
}

extern "C" void kernel_launch(void* const* d_in, const int* in_sizes, int n_in,
                              void* d_out, int out_size, void* d_ws, size_t ws_size,
                              hipStream_t stream) {
    (void)n_in; (void)d_ws; (void)ws_size; (void)out_size;

    const float* x = (const float*)d_in[0];
    float* out     = (float*)d_out;

    // total elements = T * B * C * H * W ; one float4 per thread-iteration
    const int total      = in_sizes[0];                 // 33,554,432
    const int nvec_per_t = total / (T_STEPS * 4);       // 2,097,152

    // 2048 blocks x 256 threads (8 wave32 per block) = 524,288 threads,
    // 4 grid-stride iterations each -> prefetch covers iterations 2..4.
    dim3 grid(2048), block(256);
    lif_stream_kernel<<<grid, block, 0, stream>>>(x, out, nvec_per_t);
}